// SCQPTHNet_15487652069762
// MI455X (gfx1250) — compile-verified
//
#include <hip/hip_runtime.h>
#include <hip/hip_bf16.h>
#include <math.h>

typedef __attribute__((ext_vector_type(2))) float v2f;
typedef __attribute__((ext_vector_type(8))) float v8f;

#define N 512
#define SIGMA 1e-6f
#define N_ADMM 50

// ---------------------------------------------------------------------------
// Kernel 1: C = A^T * A per batch via V_WMMA_F32_16X16X4_F32.
// Each wave computes a 32x32 output tile as 2x2 WMMA tiles (register tiling:
// 4 fragment loads feed 4 WMMAs -> 2x the flops/byte of a single-tile wave).
// 8 waves/block arranged 2(M) x 4(N) -> block tile 64(M) x 128(N).
// A-operand element (m,k) = A[k][m]; B-operand element (k,n) = A[k][n].
// ---------------------------------------------------------------------------
__global__ __launch_bounds__(256) void ata_wmma_kernel(const float* __restrict__ A,
                                                       float* __restrict__ C) {
  const int b = blockIdx.z;
  const float* Ab = A + (size_t)b * N * N;
  float* Cb = C + (size_t)b * N * N;

  const int wave = threadIdx.x >> 5;   // 0..7
  const int lane = threadIdx.x & 31;
  const int wm = wave & 1;             // 2 waves along M
  const int wn = wave >> 1;            // 4 waves along N
  const int m0 = blockIdx.y * 64 + wm * 32;
  const int n0 = blockIdx.x * 128 + wn * 32;
  const int l15 = lane & 15;
  const int hi  = lane >> 4;           // 0: K=k0,k0+1 ; 1: K=k0+2,k0+3

  v8f c00 = {0.f,0.f,0.f,0.f,0.f,0.f,0.f,0.f};
  v8f c01 = c00, c10 = c00, c11 = c00;

  for (int k0 = 0; k0 < N; k0 += 4) {
    const float* r0 = Ab + (size_t)(k0 + hi * 2) * N;
    const float* r1 = r0 + N;
    v2f a0, a1, b0, b1;
    a0.x = r0[m0 + l15];       a0.y = r1[m0 + l15];
    a1.x = r0[m0 + 16 + l15];  a1.y = r1[m0 + 16 + l15];
    b0.x = r0[n0 + l15];       b0.y = r1[n0 + l15];
    b1.x = r0[n0 + 16 + l15];  b1.y = r1[n0 + 16 + l15];
    c00 = __builtin_amdgcn_wmma_f32_16x16x4_f32(false, a0, false, b0, (short)0, c00, false, false);
    c01 = __builtin_amdgcn_wmma_f32_16x16x4_f32(false, a0, false, b1, (short)0, c01, false, false);
    c10 = __builtin_amdgcn_wmma_f32_16x16x4_f32(false, a1, false, b0, (short)0, c10, false, false);
    c11 = __builtin_amdgcn_wmma_f32_16x16x4_f32(false, a1, false, b1, (short)0, c11, false, false);
  }

  // C/D layout: VGPR v -> row m+v (lanes 0-15) / m+v+8 (lanes 16-31), col n+lane%16
  const int rb = hi * 8;
#pragma unroll
  for (int v = 0; v < 8; ++v) {
    Cb[(size_t)(m0 + rb + v) * N + (n0 + l15)]           = c00[v];
    Cb[(size_t)(m0 + rb + v) * N + (n0 + 16 + l15)]      = c01[v];
    Cb[(size_t)(m0 + 16 + rb + v) * N + (n0 + l15)]      = c10[v];
    Cb[(size_t)(m0 + 16 + rb + v) * N + (n0 + 16 + l15)] = c11[v];
  }
}

// ---------------------------------------------------------------------------
// Kernel 2: per-batch rho = clip(||Q||_F / ||ATA||_F, 1e-6, 1e6)  ((nA/nx)^.5==1)
// ---------------------------------------------------------------------------
__global__ __launch_bounds__(256) void rho_kernel(const float* __restrict__ Q,
                                                  const float* __restrict__ ATA,
                                                  float* __restrict__ rho) {
  const int b = blockIdx.x;
  const float* Qb = Q + (size_t)b * N * N;
  const float* Ab = ATA + (size_t)b * N * N;
  __shared__ float sQ[256];
  __shared__ float sA[256];
  float aq = 0.f, aa = 0.f;
  for (int i = threadIdx.x; i < N * N; i += 256) {
    float q = Qb[i]; aq += q * q;
    float a = Ab[i]; aa += a * a;
  }
  sQ[threadIdx.x] = aq;
  sA[threadIdx.x] = aa;
  __syncthreads();
  for (int s = 128; s > 0; s >>= 1) {
    if (threadIdx.x < s) {
      sQ[threadIdx.x] += sQ[threadIdx.x + s];
      sA[threadIdx.x] += sA[threadIdx.x + s];
    }
    __syncthreads();
  }
  if (threadIdx.x == 0) {
    float r = sqrtf(sQ[0]) / sqrtf(sA[0]);
    r = fminf(fmaxf(r, 1e-6f), 1e6f);
    rho[b] = r;
  }
}

// ---------------------------------------------------------------------------
// Kernel 3: M = Q + rho*ATA + sigma*I  (in place on the ATA buffer)
// ---------------------------------------------------------------------------
__global__ __launch_bounds__(256) void form_m_kernel(const float* __restrict__ Q,
                                                     float* __restrict__ M,
                                                     const float* __restrict__ rho) {
  const int b = blockIdx.y;
  const float r = rho[b];
  const size_t base = (size_t)b * N * N;
  const int idx = blockIdx.x * 256 + threadIdx.x;   // < N*N
  float v = Q[base + idx] + r * M[base + idx];
  if ((idx / N) == (idx % N)) v += SIGMA;
  M[base + idx] = v;
}

// ---------------------------------------------------------------------------
// Kernel 4: in-place Cholesky of M, one workgroup per batch (right-looking,
// pivot column staged in LDS). Each scaled column is ALSO mirrored into the
// (otherwise unused) upper triangle, so the buffer ends up holding both L
// (lower) and L^T (upper). All later solve accesses become row reads.
// ---------------------------------------------------------------------------
__global__ __launch_bounds__(256) void chol_kernel(float* __restrict__ M) {
  const int b = blockIdx.x;
  float* Mb = M + (size_t)b * N * N;
  __shared__ float col[N];
  __shared__ float sdiag;
  const int tid = threadIdx.x;

  for (int j = 0; j < N; ++j) {
    if (tid == 0) {
      float d = sqrtf(Mb[(size_t)j * N + j]);
      Mb[(size_t)j * N + j] = d;
      sdiag = d;
    }
    __syncthreads();
    const float inv = 1.0f / sdiag;
    for (int i = j + 1 + tid; i < N; i += 256) {
      float l = Mb[(size_t)i * N + j] * inv;
      Mb[(size_t)i * N + j] = l;     // L (lower)
      Mb[(size_t)j * N + i] = l;     // L^T mirror (upper, coalesced row write)
      col[i] = l;
    }
    __syncthreads();
    // trailing rank-1 update, lower triangle only (upper mirror untouched)
    for (int i = j + 1 + tid; i < N; i += 256) {
      const float li = col[i];
      float* rowp = Mb + (size_t)i * N;
      for (int k = j + 1; k <= i; ++k) rowp[k] -= li * col[k];
    }
    __syncthreads();
  }
}

// ---------------------------------------------------------------------------
// Kernel 5: 50 ADMM iterations, one workgroup (256 thr / 8 waves) per batch.
// State vectors live in LDS. Triangular solves are blocked by 16:
//   - 16x16 diagonal block solved wave-synchronously by wave 0 via __shfl
//     broadcasts (wave32, no barriers inside),
//   - rank-16 panel update by all 256 threads, fully coalesced row reads
//     thanks to the L/L^T mirrored buffer: every access is M[(base+jj)*N + i].
// Barrier rounds per iteration: ~64 (was ~1024 with column-at-a-time solves).
// ---------------------------------------------------------------------------
__global__ __launch_bounds__(256) void admm_kernel(const float* __restrict__ A,
                                                   const float* __restrict__ L,
                                                   const float* __restrict__ p,
                                                   const float* __restrict__ lb,
                                                   const float* __restrict__ ub,
                                                   const float* __restrict__ rho_arr,
                                                   float* __restrict__ xout) {
  const int b = blockIdx.x;
  const float* Ab = A + (size_t)b * N * N;
  const float* Lb = L + (size_t)b * N * N;

  __shared__ float sx[N], sz[N], su[N], sy[N], st[N];
  __shared__ float sp[N], slb[N], sub_[N];

  const int tid = threadIdx.x;
  const int lane = tid & 31;
  const int wave = tid >> 5;
  const int l15 = lane & 15;
  const float rho = rho_arr[b];

  for (int i = tid; i < N; i += 256) {
    sx[i] = 0.f; sz[i] = 0.f; su[i] = 0.f;
    sp[i]   = p[(size_t)b * N + i];
    slb[i]  = lb[(size_t)b * N + i];
    sub_[i] = ub[(size_t)b * N + i];
  }
  __syncthreads();

  for (int it = 0; it < N_ADMM; ++it) {
    // t = z - u
    for (int i = tid; i < N; i += 256) st[i] = sz[i] - su[i];
    __syncthreads();

    // rhs = -p + rho * A^T t + sigma * x   (column of A == coalesced lanes)
    for (int i = tid; i < N; i += 256) {
      float acc = 0.f;
      for (int k = 0; k < N; ++k) acc += Ab[(size_t)k * N + i] * st[k];
      sy[i] = -sp[i] + rho * acc + SIGMA * sx[i];
    }
    __syncthreads();

    // ---- forward solve: L w = rhs (in place in sy), blocked by 16 ----
    for (int jb = 0; jb < N / 16; ++jb) {
      const int base = jb * 16;
      if (wave == 0) {
        float lv[16];
#pragma unroll
        for (int jj = 0; jj < 16; ++jj)
          lv[jj] = Lb[(size_t)(base + jj) * N + (base + l15)]; // upper: L[base+l15][base+jj]
        const float ldiag = lv[l15];
        float v = sy[base + l15];
#pragma unroll
        for (int jj = 0; jj < 16; ++jj) {
          const float yj = __shfl(v, jj, 32) / __shfl(ldiag, jj, 32);
          if (l15 == jj) v = yj;
          else if (l15 > jj) v -= lv[jj] * yj;
        }
        if (lane < 16) sy[base + lane] = v;
      }
      __syncthreads();
      float yblk[16];
#pragma unroll
      for (int jj = 0; jj < 16; ++jj) yblk[jj] = sy[base + jj];
      for (int i = base + 16 + tid; i < N; i += 256) {
        float acc = sy[i];
#pragma unroll
        for (int jj = 0; jj < 16; ++jj)
          acc -= Lb[(size_t)(base + jj) * N + i] * yblk[jj];  // upper: L[i][base+jj]
        sy[i] = acc;
      }
      __syncthreads();
    }

    // ---- backward solve: L^T x = w (in place in sy), blocked by 16 ----
    for (int jb = N / 16 - 1; jb >= 0; --jb) {
      const int base = jb * 16;
      if (wave == 0) {
        float lv[16];
#pragma unroll
        for (int jj = 0; jj < 16; ++jj)
          lv[jj] = Lb[(size_t)(base + jj) * N + (base + l15)]; // lower: L^T[base+l15][base+jj]
        const float ldiag = lv[l15];
        float v = sy[base + l15];
#pragma unroll
        for (int jj = 15; jj >= 0; --jj) {
          const float xj = __shfl(v, jj, 32) / __shfl(ldiag, jj, 32);
          if (l15 == jj) v = xj;
          else if (l15 < jj) v -= lv[jj] * xj;
        }
        if (lane < 16) sy[base + lane] = v;
      }
      __syncthreads();
      float xblk[16];
#pragma unroll
      for (int jj = 0; jj < 16; ++jj) xblk[jj] = sy[base + jj];
      for (int i = tid; i < base; i += 256) {
        float acc = sy[i];
#pragma unroll
        for (int jj = 0; jj < 16; ++jj)
          acc -= Lb[(size_t)(base + jj) * N + i] * xblk[jj];  // lower: L^T[i][base+jj]
        sy[i] = acc;
      }
      __syncthreads();
    }

    for (int i = tid; i < N; i += 256) sx[i] = sy[i];
    __syncthreads();

    // Ax: 8 waves, each wave owns rows; lanes stride K then shuffle-reduce
    for (int row = wave; row < N; row += 8) {
      const float* rp = Ab + (size_t)row * N;
      float acc = 0.f;
      for (int k = lane; k < N; k += 32) acc += rp[k] * sx[k];
      for (int off = 16; off > 0; off >>= 1) acc += __shfl_down(acc, off, 32);
      if (lane == 0) st[row] = acc;
    }
    __syncthreads();

    // z = clip(Ax + u, lb, ub);  u += Ax - z
    for (int i = tid; i < N; i += 256) {
      const float ax = st[i];
      float z = fminf(fmaxf(ax + su[i], slb[i]), sub_[i]);
      su[i] = su[i] + ax - z;
      sz[i] = z;
    }
    __syncthreads();
  }

  for (int i = tid; i < N; i += 256) xout[(size_t)b * N + i] = sx[i];
}

// ---------------------------------------------------------------------------
extern "C" void kernel_launch(void* const* d_in, const int* in_sizes, int n_in,
                              void* d_out, int out_size, void* d_ws, size_t ws_size,
                              hipStream_t stream) {
  const float* Q  = (const float*)d_in[0];
  const float* p  = (const float*)d_in[1];
  const float* A  = (const float*)d_in[2];
  const float* lb = (const float*)d_in[3];
  const float* ub = (const float*)d_in[4];
  const int B = in_sizes[0] / (N * N);   // 128

  // Workspace: one B*N*N buffer reused as ATA -> M -> {L | L^T}, plus rho[B].
  float* M   = (float*)d_ws;
  float* rho = M + (size_t)B * N * N;

  dim3 g1(N / 128, N / 64, B);
  ata_wmma_kernel<<<g1, 256, 0, stream>>>(A, M);

  rho_kernel<<<B, 256, 0, stream>>>(Q, M, rho);

  dim3 g3((N * N) / 256, B);
  form_m_kernel<<<g3, 256, 0, stream>>>(Q, M, rho);

  chol_kernel<<<B, 256, 0, stream>>>(M);

  admm_kernel<<<B, 256, 0, stream>>>(A, M, p, lb, ub, rho, (float*)d_out);
}